// PaiNNInteraction_72129680769206
// MI455X (gfx1250) — compile-verified
//
#include <hip/hip_runtime.h>

typedef __attribute__((ext_vector_type(2))) float v2f;
typedef __attribute__((ext_vector_type(8))) float v8f;

constexpr int kF  = 64;    // n_atom_basis
constexpr int kTF = 192;   // 3*F
constexpr int kHLds = 68;  // LDS row stride (floats) to avoid bank conflicts

// Fast silu: v * sigmoid(v) using v_exp_f32 + v_rcp_f32 (no IEEE divide seq).
__device__ __forceinline__ float fast_silu(float v) {
  return v * __builtin_amdgcn_rcpf(1.0f + __expf(-v));
}

// ---------------------------------------------------------------------------
// Kernel 1: per-atom MLP  x = silu(q @ W1 + b1) @ W2 + b2  via fp32 WMMA.
// One workgroup = 16 atoms, 4 waves. GEMM1: each wave owns one 16-col tile of
// h (64 = 4 tiles). GEMM2: each wave owns 3 of the 12 16-col tiles of x (192).
// WMMA shape 16x16x4 fp32 -> 16 K-steps over K=64.
// ---------------------------------------------------------------------------
__global__ __launch_bounds__(128) void painn_mlp_wmma(
    const float* __restrict__ q, const float* __restrict__ W1,
    const float* __restrict__ b1, const float* __restrict__ W2,
    const float* __restrict__ b2, float* __restrict__ x, int n_atoms) {
  __shared__ float h_lds[16 * kHLds];

  const int lane = threadIdx.x & 31;
  const int wave = threadIdx.x >> 5;
  const int half = lane >> 4;   // 0: lanes 0-15, 1: lanes 16-31
  const int col  = lane & 15;
  const int atom0 = blockIdx.x * 16;

  // Clamped row index for tail safety (loads only; WMMA keeps full EXEC).
  int arow = atom0 + col;
  if (arow >= n_atoms) arow = n_atoms - 1;

  // ---------------- GEMM1: h = silu(q[16x64] @ W1[64x64] + b1) -------------
  {
    const int n = wave * 16 + col;
    const float bias = b1[n];
    v8f c;
#pragma unroll
    for (int i = 0; i < 8; ++i) c[i] = bias;

#pragma unroll
    for (int kb = 0; kb < 16; ++kb) {
      const int k0 = kb * 4 + 2 * half;  // A/B VGPR0 holds K=k0, VGPR1 K=k0+1
      v2f a = *(const v2f*)(q + (size_t)arow * kF + k0);
      v2f b;
      b.x = W1[(k0 + 0) * kF + n];
      b.y = W1[(k0 + 1) * kF + n];
      c = __builtin_amdgcn_wmma_f32_16x16x4_f32(false, a, false, b, (short)0, c,
                                                false, false);
    }
    // silu and stage to LDS; C layout: VGPR i -> row (i + 8*half), col n.
#pragma unroll
    for (int i = 0; i < 8; ++i) {
      h_lds[(i + 8 * half) * kHLds + wave * 16 + col] = fast_silu(c[i]);
    }
  }
  __syncthreads();

  // ---------------- GEMM2: x = h[16x64] @ W2[64x192] + b2 ------------------
  // Preload all 16 A fragments from LDS once (reused across 3 N-tiles).
  v2f afrag[16];
#pragma unroll
  for (int kb = 0; kb < 16; ++kb) {
    const int k0 = kb * 4 + 2 * half;
    afrag[kb] = *(const v2f*)(&h_lds[col * kHLds + k0]);
  }

#pragma unroll
  for (int t = 0; t < 3; ++t) {
    const int n = (wave + 4 * t) * 16 + col;
    const float bias = b2[n];
    v8f c;
#pragma unroll
    for (int i = 0; i < 8; ++i) c[i] = bias;

#pragma unroll
    for (int kb = 0; kb < 16; ++kb) {
      const int k0 = kb * 4 + 2 * half;
      v2f b;
      b.x = W2[(k0 + 0) * kTF + n];
      b.y = W2[(k0 + 1) * kTF + n];
      c = __builtin_amdgcn_wmma_f32_16x16x4_f32(false, afrag[kb], false, b,
                                                (short)0, c, false, false);
    }
#pragma unroll
    for (int i = 0; i < 8; ++i) {
      const int row = atom0 + i + 8 * half;
      if (row < n_atoms) x[(size_t)row * kTF + n] = c[i];
    }
  }
}

// ---------------------------------------------------------------------------
// Kernel 2: edge phase. One wave (32 lanes) per edge; lane owns 2 features.
// Wij is streamed once -> non-temporal loads. x[j]/mu[j] gathers hit L2
// (38 MB each, L2 = 192 MB). Scatter-add with global_atomic_add_f32 into
// d_out, which was pre-initialized to (q, mu).
// ---------------------------------------------------------------------------
__global__ __launch_bounds__(256) void painn_edges(
    const float* __restrict__ Wij, const float* __restrict__ dir_ij,
    const int* __restrict__ idx_i, const int* __restrict__ idx_j,
    const float* __restrict__ x, const float* __restrict__ mu,
    float* __restrict__ out_q, float* __restrict__ out_mu, int n_edges) {
  const int lane = threadIdx.x & 31;
  const int edge = blockIdx.x * 8 + (threadIdx.x >> 5);
  if (edge >= n_edges) return;

  const int j = idx_j[edge];
  const int i = idx_i[edge];
  const int f = lane * 2;

  const float* w = Wij + (size_t)edge * kTF;
  // Pull the next block's Wij cacheline toward us.
  __builtin_prefetch(w + (size_t)8 * kTF, 0, 0);

  // Streamed weights: non-temporal (don't pollute L2 gather working set).
  const v2f wq = __builtin_nontemporal_load((const v2f*)(w + f));
  const v2f wr = __builtin_nontemporal_load((const v2f*)(w + kF + f));
  const v2f wm = __builtin_nontemporal_load((const v2f*)(w + 2 * kF + f));

  const float* xj = x + (size_t)j * kTF;
  const v2f xq = *(const v2f*)(xj + f);
  const v2f xr = *(const v2f*)(xj + kF + f);
  const v2f xm = *(const v2f*)(xj + 2 * kF + f);

  const float* muj = mu + (size_t)j * kTF;
  const v2f m0 = *(const v2f*)(muj + f);
  const v2f m1 = *(const v2f*)(muj + kF + f);
  const v2f m2 = *(const v2f*)(muj + 2 * kF + f);

  const float d0 = dir_ij[edge * 3 + 0];
  const float d1 = dir_ij[edge * 3 + 1];
  const float d2 = dir_ij[edge * 3 + 2];

  const v2f dq = wq * xq;   // dq_e
  const v2f dr = wr * xr;   // dmuR
  const v2f dm = wm * xm;   // dmumu

  float* oq = out_q + (size_t)i * kF + f;
  atomicAdd(oq + 0, dq.x);
  atomicAdd(oq + 1, dq.y);

  float* om = out_mu + (size_t)i * kTF + f;
  const v2f u0 = dr * d0 + dm * m0;
  atomicAdd(om + 0, u0.x);
  atomicAdd(om + 1, u0.y);
  const v2f u1 = dr * d1 + dm * m1;
  atomicAdd(om + kF + 0, u1.x);
  atomicAdd(om + kF + 1, u1.y);
  const v2f u2 = dr * d2 + dm * m2;
  atomicAdd(om + 2 * kF + 0, u2.x);
  atomicAdd(om + 2 * kF + 1, u2.y);
}

extern "C" void kernel_launch(void* const* d_in, const int* in_sizes, int n_in,
                              void* d_out, int out_size, void* d_ws,
                              size_t ws_size, hipStream_t stream) {
  const float* q      = (const float*)d_in[0];
  const float* mu     = (const float*)d_in[1];
  const float* Wij    = (const float*)d_in[2];
  const float* dir_ij = (const float*)d_in[3];
  const int*   idx_i  = (const int*)d_in[4];
  const int*   idx_j  = (const int*)d_in[5];
  // d_in[6] = n_atoms scalar (derive from sizes instead)
  const float* W1     = (const float*)d_in[7];
  const float* b1     = (const float*)d_in[8];
  const float* W2     = (const float*)d_in[9];
  const float* b2     = (const float*)d_in[10];

  const int N = in_sizes[0] / kF;  // n_atoms
  const int E = in_sizes[4];       // n_edges (idx_i count)

  float* out    = (float*)d_out;
  float* out_q  = out;                        // [N, 64]
  float* out_mu = out + (size_t)N * kF;       // [N, 3, 64]
  float* x = (float*)d_ws;                    // [N, 192] scratch

  // Residual base: out = (q, mu). Graph-capture-safe d2d async copies.
  hipMemcpyAsync(out_q, q, (size_t)N * kF * sizeof(float),
                 hipMemcpyDeviceToDevice, stream);
  hipMemcpyAsync(out_mu, mu, (size_t)N * kTF * sizeof(float),
                 hipMemcpyDeviceToDevice, stream);

  // MLP: 16 atoms per workgroup, 4 waves of 32 (wave32).
  painn_mlp_wmma<<<(N + 15) / 16, 128, 0, stream>>>(q, W1, b1, W2, b2, x, N);

  // Edges: 8 edges per 256-thread block (one wave32 per edge).
  painn_edges<<<(E + 7) / 8, 256, 0, stream>>>(Wij, dir_ij, idx_i, idx_j, x,
                                               mu, out_q, out_mu, E);
}